// RelMultiHeadAttention_9466107921194
// MI455X (gfx1250) — compile-verified
//
#include <hip/hip_runtime.h>
#include <math.h>

typedef __attribute__((ext_vector_type(2))) float v2f;
typedef __attribute__((ext_vector_type(8))) float v8f;

// Explicit global-address-space pointers -> global_load_b64 (not flat_load).
typedef const float __attribute__((address_space(1)))* gcf;
typedef float __attribute__((address_space(1)))* gf;

#define LSEQ 1024
#define NHEAD 8
#define DK 128
#define RPOS 2047   // 2L-1
#define RFEAT 192
#define DMODEL 1024

static __device__ __forceinline__ v8f wmma4(v2f a, v2f b, v8f c) {
    // D = A(16x4 f32) * B(4x16 f32) + C(16x16 f32)
    return __builtin_amdgcn_wmma_f32_16x16x4_f32(
        /*neg_a=*/false, a, /*neg_b=*/false, b,
        /*c_mod=*/(short)0, c, /*reuse_a=*/false, /*reuse_b=*/false);
}

static __device__ __forceinline__ int imin(int a, int b) { return a < b ? a : b; }

// One K-step worth of operands: A fragment + 4 B fragments (16Mx64N strip).
struct Frag { v2f a, b0, b1, b2, b3; };

static __device__ __forceinline__ void load_frag(
    Frag& f, gcf& pA, gcf (&pB)[4],
    long long bo0, long long bo1, long long bstep)
{
    f.a.x  = pA[0];      f.a.y  = pA[1];      pA += 4;
    f.b0.x = pB[0][bo0]; f.b0.y = pB[0][bo1]; pB[0] += bstep;
    f.b1.x = pB[1][bo0]; f.b1.y = pB[1][bo1]; pB[1] += bstep;
    f.b2.x = pB[2][bo0]; f.b2.y = pB[2][bo1]; pB[2] += bstep;
    f.b3.x = pB[3][bo0]; f.b3.y = pB[3][bo1]; pB[3] += bstep;
}

static __device__ __forceinline__ void mma_frag(const Frag& f, v8f (&acc)[4])
{
    acc[0] = wmma4(f.a, f.b0, acc[0]);
    acc[1] = wmma4(f.a, f.b1, acc[1]);
    acc[2] = wmma4(f.a, f.b2, acc[2]);
    acc[3] = wmma4(f.a, f.b3, acc[3]);
}

// ---------------------------------------------------------------------------
// Generic batched WMMA fp32 GEMM:  C = A * B (or A * B^T), optional bias.
// Block = 128 threads = 4 waves; wave computes a 16(M) x 64(N) strip.
// Out-of-range rows/cols handled by CLAMPING load addresses (garbage only
// reaches never-stored outputs) -> branch-free, EXEC-stable hot loop.
// Ping-pong unroll-by-2 software pipeline (K/4 must be even, true for all
// uses: K in {1024, 192, 128}) -> no register-rotation copies.
// Per blockIdx.z: b = z / Hdiv, h = z % Hdiv; pointers offset by b*OffB+h*OffH.
// ---------------------------------------------------------------------------
template <bool BT, bool BIAS>
__global__ __launch_bounds__(128) void gemm_f32_wmma(
    const float* __restrict__ A, int lda, long long aOffB, long long aOffH,
    const float* __restrict__ B, int ldb, long long bOffB, long long bOffH,
    float* __restrict__ C, int ldc, long long cOffB, long long cOffH,
    int M, int N, int K, int Hdiv, const float* __restrict__ bias)
{
    const int z  = blockIdx.z;
    const int bb = z / Hdiv, hh = z % Hdiv;
    A += (long long)bb * aOffB + (long long)hh * aOffH;
    B += (long long)bb * bOffB + (long long)hh * bOffH;
    C += (long long)bb * cOffB + (long long)hh * cOffH;

    const int lane = threadIdx.x & 31;
    const int wave = threadIdx.x >> 5;
    const int g    = lane >> 4;       // lane group (0: lanes 0-15, 1: 16-31)
    const int l15  = lane & 15;
    const int m0   = blockIdx.y * 64 + wave * 16;
    const int n0   = blockIdx.x * 64;

    // Clamped A row pointer (garbage rows never stored).
    const int rowAc = imin(m0 + l15, M - 1);
    const float* pA0 = A + (long long)rowAc * lda + 2 * g;

    // Clamped B column pointers.
    const float* pB0[4];
#pragma unroll
    for (int j = 0; j < 4; ++j) {
        const int colc = imin(n0 + j * 16 + l15, N - 1);
        if (BT) pB0[j] = B + (long long)colc * ldb + 2 * g;
        else    pB0[j] = B + colc;
    }
    const long long bo0   = BT ? 0 : (long long)(2 * g) * ldb;  // NN: row 2g
    const long long bo1   = BT ? 1 : bo0 + ldb;                 // NN: row 2g+1
    const long long bstep = BT ? 4 : (long long)4 * ldb;

    __builtin_prefetch(pA0, 0, 3);
#pragma unroll
    for (int j = 0; j < 4; ++j) __builtin_prefetch(pB0[j], 0, 3);

    gcf pA = (gcf)pA0;
    gcf pB[4];
#pragma unroll
    for (int j = 0; j < 4; ++j) pB[j] = (gcf)pB0[j];

    v8f acc[4] = {v8f{}, v8f{}, v8f{}, v8f{}};

    // Ping-pong pipeline over steps = K/4 (even, >= 2).
    const int steps = K >> 2;
    Frag f, h;
    load_frag(f, pA, pB, bo0, bo1, bstep);
    for (int s = 1; s + 1 < steps; s += 2) {
        load_frag(h, pA, pB, bo0, bo1, bstep);
        mma_frag(f, acc);
        load_frag(f, pA, pB, bo0, bo1, bstep);
        mma_frag(h, acc);
    }
    load_frag(h, pA, pB, bo0, bo1, bstep);   // final (steps-1) step
    mma_frag(f, acc);
    mma_frag(h, acc);

    // --- store with real-index guards ---
#pragma unroll
    for (int j = 0; j < 4; ++j) {
#pragma unroll
        for (int v = 0; v < 8; ++v) {
            const int row = m0 + v + 8 * g;
            const int col = n0 + j * 16 + l15;
            if (row < M && col < N) {
                float o = acc[j][v];
                if (BIAS) o += bias[col];
                ((gf)C)[(long long)row * ldc + col] = o;
            }
        }
    }
}

// ---------------------------------------------------------------------------
// Positional scores with fused relative_shift:
//   P[q,j] = (Q*scale + bp)[b,q,h,:] . rel_k[j, h,:]     (NT GEMM, K=128)
//   S[b,h,q, j-1023+q] += P[q,j]   (bijective scatter -> race-free RMW)
// Same clamped-address ping-pong pipeline (steps = 32, even).
// ---------------------------------------------------------------------------
__global__ __launch_bounds__(128) void pos_scores_shift_add(
    const float* __restrict__ A2, const float* __restrict__ relk,
    float* __restrict__ S)
{
    const int z  = blockIdx.z;              // b*H + h
    const int bb = z / NHEAD, hh = z % NHEAD;
    const float* A  = A2   + (long long)bb * LSEQ * DMODEL + hh * DK;
    const float* Bt = relk + hh * DK;
    float* Sp = S + (long long)z * LSEQ * LSEQ;

    const int lane = threadIdx.x & 31;
    const int wave = threadIdx.x >> 5;
    const int g    = lane >> 4;
    const int l15  = lane & 15;
    const int m0   = blockIdx.y * 64 + wave * 16;   // q tile
    const int n0   = blockIdx.x * 64;               // j tile (0..2046)

    const float* pA0 = A + (long long)(m0 + l15) * DMODEL + 2 * g;
    const float* pB0[4];
#pragma unroll
    for (int j = 0; j < 4; ++j) {
        const int colc = imin(n0 + j * 16 + l15, RPOS - 1);
        pB0[j] = Bt + (long long)colc * DMODEL + 2 * g;
    }

    __builtin_prefetch(pA0, 0, 3);
#pragma unroll
    for (int j = 0; j < 4; ++j) __builtin_prefetch(pB0[j], 0, 3);

    gcf pA = (gcf)pA0;
    gcf pB[4];
#pragma unroll
    for (int j = 0; j < 4; ++j) pB[j] = (gcf)pB0[j];

    v8f acc[4] = {v8f{}, v8f{}, v8f{}, v8f{}};

    const int steps = DK >> 2;               // 32
    Frag f, h;
    load_frag(f, pA, pB, 0, 1, 4);
    for (int s = 1; s + 1 < steps; s += 2) {
        load_frag(h, pA, pB, 0, 1, 4);
        mma_frag(f, acc);
        load_frag(f, pA, pB, 0, 1, 4);
        mma_frag(h, acc);
    }
    load_frag(h, pA, pB, 0, 1, 4);
    mma_frag(f, acc);
    mma_frag(h, acc);

#pragma unroll
    for (int j = 0; j < 4; ++j) {
#pragma unroll
        for (int v = 0; v < 8; ++v) {
            const int q  = m0 + v + 8 * g;
            const int jj = n0 + j * 16 + l15;
            const int kk = jj - (LSEQ - 1) + q;     // relative_shift gather index
            if (jj < RPOS && kk >= 0 && kk < LSEQ) {
                gf p = (gf)(Sp + (long long)q * LSEQ + kk);
                *p += acc[j][v];
            }
        }
    }
}

// ---------------------------------------------------------------------------
// Positional embedding (2047 x 192). One wave32 per distance row; lane = n
// index (n = 32 == wave width). Gamma-feature row max via __shfl_xor.
// ---------------------------------------------------------------------------
__global__ __launch_bounds__(32) void pos_embed_kernel(float* __restrict__ pos)
{
    const int i = blockIdx.x;        // 0..2046
    const int j = threadIdx.x;       // 0..31  (n = feature_size/6 = 32)
    const float dist = (float)(i - (LSEQ - 1));
    const float absd = fabsf(dist);

    // exponential features: half_life = 2^linspace(3, log2(1024)=10, 32)
    const float hl = exp2f(3.0f + 7.0f * (float)j / 31.0f);
    const float fe = expf(-0.6931471805599453f / hl * absd);

    // central-mask features: cw = 2^(j+1) - 1
    const float cw = exp2f((float)(j + 1)) - 1.0f;
    const float cm = (cw > absd) ? 1.0f : 0.0f;

    // gamma features: mean = linspace(32,1024,32) = 32*(j+1); sd = 16
    const float mean = 32.0f * (float)(j + 1);
    const float conc = (mean / 16.0f) * (mean / 16.0f);
    const float rate = mean / 256.0f;
    float lu = ((absd > 0.0f) ? (conc - 1.0f) * logf(absd) : -INFINITY)
               - rate * absd;
    const float ln_ = lgammaf(conc) - conc * logf(rate);
    float prob = expf(lu - ln_) + 1e-8f;

    float mx = prob;
#pragma unroll
    for (int o = 16; o; o >>= 1) mx = fmaxf(mx, __shfl_xor(mx, o, 32));
    const float fg = prob / mx;

    const float s = dist > 0.0f ? 1.0f : (dist < 0.0f ? -1.0f : 0.0f);
    float* r = pos + (long long)i * RFEAT;
    r[j]        = fe;     r[32 + j]  = cm;     r[64 + j]  = fg;
    r[96 + j]   = s * fe; r[128 + j] = s * cm; r[160 + j] = s * fg;
}

// ---------------------------------------------------------------------------
// A1 = Q*scale + rel_embed_bias ; A2 = Q*scale + rel_pos_bias  (bias flat 1024)
// ---------------------------------------------------------------------------
__global__ __launch_bounds__(256) void make_biased_q(
    const float* __restrict__ Qp, const float* __restrict__ be,
    const float* __restrict__ bp, float* __restrict__ A1, float* __restrict__ A2)
{
    const long long idx = (long long)blockIdx.x * 256 + threadIdx.x;
    const int c = (int)(idx & (DMODEL - 1));
    const float qv = Qp[idx] * 0.08838834764831845f;  // 128^-0.5
    A1[idx] = qv + be[c];
    A2[idx] = qv + bp[c];
}

// ---------------------------------------------------------------------------
// Row softmax over 1024 columns, in place. One 256-thread block per row.
// ---------------------------------------------------------------------------
__global__ __launch_bounds__(256) void softmax_rows(float* __restrict__ S)
{
    float* p = S + (long long)blockIdx.x * LSEQ;
    const int tid = threadIdx.x;
    const int lane = tid & 31, wv = tid >> 5;
    __shared__ float sm[8], ss[8];

    float v[4], m = -INFINITY;
#pragma unroll
    for (int i = 0; i < 4; ++i) { v[i] = p[tid + i * 256]; m = fmaxf(m, v[i]); }
#pragma unroll
    for (int o = 16; o; o >>= 1) m = fmaxf(m, __shfl_xor(m, o, 32));
    if (lane == 0) sm[wv] = m;
    __syncthreads();
    if (tid == 0) {
        float t = sm[0];
#pragma unroll
        for (int i = 1; i < 8; ++i) t = fmaxf(t, sm[i]);
        sm[0] = t;
    }
    __syncthreads();
    m = sm[0];

    float sum = 0.0f;
#pragma unroll
    for (int i = 0; i < 4; ++i) { v[i] = expf(v[i] - m); sum += v[i]; }
#pragma unroll
    for (int o = 16; o; o >>= 1) sum += __shfl_xor(sum, o, 32);
    if (lane == 0) ss[wv] = sum;
    __syncthreads();
    if (tid == 0) {
        float t = 0.0f;
#pragma unroll
        for (int i = 0; i < 8; ++i) t += ss[i];
        ss[0] = t;
    }
    __syncthreads();
    const float inv = 1.0f / ss[0];
#pragma unroll
    for (int i = 0; i < 4; ++i) p[tid + i * 256] = v[i] * inv;
}

// ---------------------------------------------------------------------------
extern "C" void kernel_launch(void* const* d_in, const int* in_sizes, int n_in,
                              void* d_out, int out_size, void* d_ws, size_t ws_size,
                              hipStream_t stream)
{
    const float* q     = (const float*)d_in[0];  // (2,1024,1024)
    const float* w_qs  = (const float*)d_in[1];  // (1024,1024)
    const float* w_ks  = (const float*)d_in[2];
    const float* w_vs  = (const float*)d_in[3];
    const float* w_rel = (const float*)d_in[4];  // (192,1024)
    const float* be    = (const float*)d_in[5];  // (1,8,1,128) -> 1024
    const float* bp    = (const float*)d_in[6];
    const float* w_out = (const float*)d_in[7];  // (1024,1024)
    const float* b_out = (const float*)d_in[8];  // (1024,)
    float* out = (float*)d_out;

    float* ws = (float*)d_ws;
    const long long NQK = 2048LL * DMODEL;
    float* Qp   = ws; ws += NQK;
    float* Kp   = ws; ws += NQK;
    float* Vp   = ws; ws += NQK;
    float* A1   = ws; ws += NQK;
    float* A2   = ws; ws += NQK;
    float* relk = ws; ws += NQK;            // 2047 rows used, padded to 2048
    float* ctx  = ws; ws += NQK;
    float* pos  = ws; ws += 2048LL * RFEAT;
    float* S    = ws; ws += 16LL * 1024 * 1024;  // (B,H,L,L)

    const dim3 blk(128);
    const long long M1 = 1024LL * 1024;

    // Q/K/V projections: (2048,1024) @ (1024,1024)
    gemm_f32_wmma<false, false><<<dim3(16, 32, 1), blk, 0, stream>>>(
        q, DMODEL, 0, 0, w_qs, DMODEL, 0, 0, Qp, DMODEL, 0, 0,
        2048, DMODEL, DMODEL, 1, nullptr);
    gemm_f32_wmma<false, false><<<dim3(16, 32, 1), blk, 0, stream>>>(
        q, DMODEL, 0, 0, w_ks, DMODEL, 0, 0, Kp, DMODEL, 0, 0,
        2048, DMODEL, DMODEL, 1, nullptr);
    gemm_f32_wmma<false, false><<<dim3(16, 32, 1), blk, 0, stream>>>(
        q, DMODEL, 0, 0, w_vs, DMODEL, 0, 0, Vp, DMODEL, 0, 0,
        2048, DMODEL, DMODEL, 1, nullptr);

    // positional embeddings (2047,192), then rel_k = pos @ w_rel (2047,1024)
    pos_embed_kernel<<<RPOS, 32, 0, stream>>>(pos);
    gemm_f32_wmma<false, false><<<dim3(16, 32, 1), blk, 0, stream>>>(
        pos, RFEAT, 0, 0, w_rel, DMODEL, 0, 0, relk, DMODEL, 0, 0,
        RPOS, DMODEL, RFEAT, 1, nullptr);

    // biased queries
    make_biased_q<<<(2048 * DMODEL) / 256, 256, 0, stream>>>(Qp, be, bp, A1, A2);

    // content scores: S[b,h,q,k] = A1[b,q,h,:] . K[b,k,h,:]   (NT, K=128)
    gemm_f32_wmma<true, false><<<dim3(16, 16, 16), blk, 0, stream>>>(
        A1, DMODEL, M1, DK, Kp, DMODEL, M1, DK, S, LSEQ, 8 * M1, M1,
        LSEQ, LSEQ, DK, NHEAD, nullptr);

    // positional scores with fused relative_shift scatter-add
    pos_scores_shift_add<<<dim3(32, 16, 16), blk, 0, stream>>>(A2, relk, S);

    // softmax over k
    softmax_rows<<<16384, 256, 0, stream>>>(S);

    // context: ctx[b,q,h,:] = softmax(S)[b,h,q,:] @ V[b,:,h,:]   (NN, N=128)
    gemm_f32_wmma<false, false><<<dim3(2, 16, 16), blk, 0, stream>>>(
        S, LSEQ, 8 * M1, M1, Vp, DMODEL, M1, DK, ctx, DMODEL, M1, DK,
        LSEQ, DK, LSEQ, NHEAD, nullptr);

    // output projection + bias: (2048,1024) @ (1024,1024) + b_out
    gemm_f32_wmma<false, true><<<dim3(16, 32, 1), blk, 0, stream>>>(
        ctx, DMODEL, 0, 0, w_out, DMODEL, 0, 0, out, DMODEL, 0, 0,
        2048, DMODEL, DMODEL, 1, b_out);
}